// AdditivetionAttention_31576599560219
// MI455X (gfx1250) — compile-verified
//
#include <hip/hip_runtime.h>
#include <hip/hip_bf16.h>

typedef float v2f __attribute__((ext_vector_type(2)));
typedef float v8f __attribute__((ext_vector_type(8)));

#if __has_builtin(__builtin_amdgcn_tanhf)
__device__ __forceinline__ float fast_tanh(float x) {
    return __builtin_amdgcn_tanhf(x);   // V_TANH_F32 (CDNA5 TRANS pipe)
}
#else
__device__ __forceinline__ float fast_tanh(float x) { return tanhf(x); }
#endif

// ---------------------------------------------------------------------------
// fp32 GEMM: C[M,N] = A[M,K] * B[K,N], row-major, batched via grid.z.
// One wave (32 threads) computes a 16(M) x 64(N) C slab as four 16x16 tiles,
// reusing the A fragment across 4 x V_WMMA_F32_16X16X4_F32 per K-step.
//
// fp32 WMMA operand layout (ISA 7.12.2, wave32):
//   A (16x4):  lane L (0..15): M=L,     v0=A[M][k+0], v1=A[M][k+1]
//              lane L (16..31): M=L-16, v0=A[M][k+2], v1=A[M][k+3]
//   B (4x16):  lane L (0..15): N=L,     v0=B[k+0][N], v1=B[k+1][N]
//              lane L (16..31): N=L-16, v0=B[k+2][N], v1=B[k+3][N]
//   C/D (16x16): VGPR i, lanes 0..15 -> (M=i,   N=lane)
//                         lanes 16..31 -> (M=8+i, N=lane-16)
// ---------------------------------------------------------------------------
__global__ __launch_bounds__(32) void gemm_f32_wmma(
    const float* __restrict__ A, const float* __restrict__ B,
    float* __restrict__ C,
    int M, int N, int K, int lda, int ldb, int ldc,
    long long sA, long long sB, long long sC)
{
    const int bz = blockIdx.z;
    A += (long long)bz * sA;
    B += (long long)bz * sB;
    C += (long long)bz * sC;

    const int tileN = blockIdx.x * 64;
    const int tileM = blockIdx.y * 16;

    const int lane = threadIdx.x;      // 0..31
    const int half = lane >> 4;        // 0 or 1 (selects K pair)
    const int l    = lane & 15;        // M row (for A) / N col (for B)

    const float* Arow = A + (long long)(tileM + l) * lda + half * 2;
    const float* Bcol = B + (long long)(half * 2) * ldb + tileN + l;

    v8f acc0 = {}, acc1 = {}, acc2 = {}, acc3 = {};

    for (int kk = 0; kk < K; kk += 4) {
        v2f a;
        a.x = Arow[kk];
        a.y = Arow[kk + 1];

        const float* b0 = Bcol + (long long)kk * ldb;
        const float* b1 = Bcol + (long long)(kk + 1) * ldb;

        v2f b;
        b.x = b0[0];  b.y = b1[0];
        acc0 = __builtin_amdgcn_wmma_f32_16x16x4_f32(false, a, false, b,
                                                     (short)0, acc0, false, false);
        b.x = b0[16]; b.y = b1[16];
        acc1 = __builtin_amdgcn_wmma_f32_16x16x4_f32(false, a, false, b,
                                                     (short)0, acc1, false, false);
        b.x = b0[32]; b.y = b1[32];
        acc2 = __builtin_amdgcn_wmma_f32_16x16x4_f32(false, a, false, b,
                                                     (short)0, acc2, false, false);
        b.x = b0[48]; b.y = b1[48];
        acc3 = __builtin_amdgcn_wmma_f32_16x16x4_f32(false, a, false, b,
                                                     (short)0, acc3, false, false);
    }

    const int row0 = tileM + half * 8;
    float* Cout = C + tileN + l;
    #pragma unroll
    for (int i = 0; i < 8; ++i) {
        float* cr = Cout + (long long)(row0 + i) * ldc;
        cr[0]  = acc0[i];
        cr[16] = acc1[i];
        cr[32] = acc2[i];
        cr[48] = acc3[i];
    }
}

// ---------------------------------------------------------------------------
// scores[b,q,k] = sum_h w_v[h] * tanh(qp[b,q,h] + kp[b,k,h])
// Block = 32 q-rows x 32 k-rows for one batch. qp/kp tiles staged in LDS with
// 129-float row stride (129 mod 64 == 1 -> conflict-free across 32 lanes).
// Thread t: ki = t&31, qb = t>>5; computes q rows {qb, qb+8, qb+16, qb+24}.
// ---------------------------------------------------------------------------
#define H_DIM   128
#define LDS_STR 129

__global__ __launch_bounds__(256) void scores_kernel(
    const float* __restrict__ qp,   // [B, 512, 128]
    const float* __restrict__ kp,   // [B, 1024, 128]
    const float* __restrict__ wv,   // [128]
    float* __restrict__ sc,         // [B, 512, 1024]
    int Q, int K)
{
    __shared__ float qs[32 * LDS_STR];
    __shared__ float ks[32 * LDS_STR];
    __shared__ float wv_s[H_DIM];

    const int b  = blockIdx.z;
    const int q0 = blockIdx.y * 32;
    const int k0 = blockIdx.x * 32;
    const int t  = threadIdx.x;

    if (t < H_DIM) wv_s[t] = wv[t];

    // Stage 32x128 tiles: 256 threads, each loads 16 floats (4x float4).
    {
        const int row = t >> 3;           // 0..31
        const int col = (t & 7) * 16;     // 0,16,...,112
        const float4* qsrc = (const float4*)(qp + ((long long)(b * Q + q0 + row)) * H_DIM + col);
        const float4* ksrc = (const float4*)(kp + ((long long)(b * K + k0 + row)) * H_DIM + col);
        #pragma unroll
        for (int v = 0; v < 4; ++v) {
            float4 qv = qsrc[v];
            float4 kv = ksrc[v];
            float* qd = &qs[row * LDS_STR + col + v * 4];
            float* kd = &ks[row * LDS_STR + col + v * 4];
            qd[0] = qv.x; qd[1] = qv.y; qd[2] = qv.z; qd[3] = qv.w;
            kd[0] = kv.x; kd[1] = kv.y; kd[2] = kv.z; kd[3] = kv.w;
        }
    }
    __syncthreads();

    const int ki = t & 31;
    const int qb = t >> 5;                 // 0..7
    const float* krow = &ks[ki * LDS_STR];

    float s0 = 0.f, s1 = 0.f, s2 = 0.f, s3 = 0.f;
    #pragma unroll 4
    for (int h = 0; h < H_DIM; ++h) {
        const float kv = krow[h];
        const float w  = wv_s[h];
        s0 += w * fast_tanh(qs[(qb     ) * LDS_STR + h] + kv);
        s1 += w * fast_tanh(qs[(qb +  8) * LDS_STR + h] + kv);
        s2 += w * fast_tanh(qs[(qb + 16) * LDS_STR + h] + kv);
        s3 += w * fast_tanh(qs[(qb + 24) * LDS_STR + h] + kv);
    }

    const long long base = ((long long)(b * Q + q0 + qb)) * K + k0 + ki;
    const long long rstep = (long long)8 * K;
    sc[base            ] = s0;
    sc[base +     rstep] = s1;
    sc[base + 2 * rstep] = s2;
    sc[base + 3 * rstep] = s3;
}

// ---------------------------------------------------------------------------
// In-place row softmax over K elements. One block (256 threads) per row.
// ---------------------------------------------------------------------------
__global__ __launch_bounds__(256) void softmax_kernel(float* __restrict__ sc, int K)
{
    __shared__ float red[256];
    float* row = sc + (long long)blockIdx.x * K;
    const int t = threadIdx.x;
    const int per = K / 256;   // 4

    float m = -3.402823466e+38f;
    #pragma unroll
    for (int i = 0; i < 4; ++i) {
        if (i < per) m = fmaxf(m, row[t + i * 256]);
    }
    red[t] = m;
    __syncthreads();
    for (int s = 128; s > 0; s >>= 1) {
        if (t < s) red[t] = fmaxf(red[t], red[t + s]);
        __syncthreads();
    }
    const float rowmax = red[0];
    __syncthreads();

    float e[4];
    float sum = 0.f;
    #pragma unroll
    for (int i = 0; i < 4; ++i) {
        if (i < per) {
            e[i] = __expf(row[t + i * 256] - rowmax);
            sum += e[i];
        }
    }
    red[t] = sum;
    __syncthreads();
    for (int s = 128; s > 0; s >>= 1) {
        if (t < s) red[t] += red[t + s];
        __syncthreads();
    }
    const float inv = 1.0f / red[0];

    #pragma unroll
    for (int i = 0; i < 4; ++i) {
        if (i < per) row[t + i * 256] = e[i] * inv;
    }
}

// ---------------------------------------------------------------------------
// Launcher
// ---------------------------------------------------------------------------
extern "C" void kernel_launch(void* const* d_in, const int* in_sizes, int n_in,
                              void* d_out, int out_size, void* d_ws, size_t ws_size,
                              hipStream_t stream)
{
    (void)in_sizes; (void)n_in; (void)out_size; (void)ws_size;

    const int B = 4, Q = 512, K = 1024, D = 512, H = 128;

    const float* queries = (const float*)d_in[0];  // [B,Q,D]
    const float* keys    = (const float*)d_in[1];  // [B,K,D]
    const float* values  = (const float*)d_in[2];  // [B,K,D]
    const float* W_q     = (const float*)d_in[3];  // [D,H]
    const float* W_k     = (const float*)d_in[4];  // [D,H]
    const float* w_v     = (const float*)d_in[5];  // [H]
    float* out = (float*)d_out;                    // [B,Q,D]

    float* qp = (float*)d_ws;                      // [B*Q, H]   1 MB
    float* kp = qp + (size_t)B * Q * H;            // [B*K, H]   2 MB
    float* sc = kp + (size_t)B * K * H;            // [B, Q, K]  8 MB

    // 1) qp = queries(2048x512) * W_q(512x128)
    {
        dim3 grid(H / 64, (B * Q) / 16, 1);
        gemm_f32_wmma<<<grid, 32, 0, stream>>>(queries, W_q, qp,
                                               B * Q, H, D, D, H, H, 0, 0, 0);
    }
    // 2) kp = keys(4096x512) * W_k(512x128)
    {
        dim3 grid(H / 64, (B * K) / 16, 1);
        gemm_f32_wmma<<<grid, 32, 0, stream>>>(keys, W_k, kp,
                                               B * K, H, D, D, H, H, 0, 0, 0);
    }
    // 3) scores = sum_h w_v * tanh(qp + kp)
    {
        dim3 grid(K / 32, Q / 32, B);
        scores_kernel<<<grid, 256, 0, stream>>>(qp, kp, w_v, sc, Q, K);
    }
    // 4) softmax over K, in place
    {
        softmax_kernel<<<B * Q, 256, 0, stream>>>(sc, K);
    }
    // 5) out = attn(512x1024) * values(1024x512), batched over B
    {
        dim3 grid(D / 64, Q / 16, B);
        gemm_f32_wmma<<<grid, 32, 0, stream>>>(sc, values, out,
                                               Q, D, K, K, D, D,
                                               (long long)Q * K,
                                               (long long)K * D,
                                               (long long)Q * D);
    }
}